// MyModel_61933428411752
// MI455X (gfx1250) — compile-verified
//
#include <hip/hip_runtime.h>

// CDNA5 / gfx1250 fused LSTM scan using V_WMMA_F32_16X16X4_F32.
//
// Per wave (32 lanes), one 16-sample batch tile, all 512 timesteps:
//   Gt[M=80 gate rows][n=16 batch cols] = W[80][32] @ V[32][16]
//   V = [ x_t (K=0..9) ; h (K=10..29) ; 0 (K=30..31) ]
// A (16x4 f32): lane L -> row M = 16*mt + (L&15); v0: K=4kk+2*(L>>4), v1: K+1
// B (4x16 f32): lane L -> col n = L&15;          v0: K=4kk+2*(L>>4), v1: K+1
// C/D (16x16 f32): lane L, vgpr v -> (M = 16*mt + v + 8*(L>>4), n = L&15)
//
// x-fragments for step t+1 are prefetched during step t (x is scan-independent),
// activations use native v_tanh_f32 when available, and all global output
// stores are LDS-staged so each store instruction is a contiguous 128B line.

typedef __attribute__((ext_vector_type(2))) float v2f;
typedef __attribute__((ext_vector_type(8))) float v8f;

#define T_STEPS 512
#define BATCH   4096
#define ISZ     10
#define HSZ     20

__device__ __forceinline__ float act_tanh(float x) {
#if __has_builtin(__builtin_amdgcn_tanhf)
    return __builtin_amdgcn_tanhf(x);          // v_tanh_f32
#else
    const float e = __expf(-2.0f * x);
    return __builtin_fmaf(-2.0f, __fdividef(e, 1.0f + e), 1.0f);
#endif
}
__device__ __forceinline__ float act_sigmoid(float x) {
    return __builtin_fmaf(0.5f, act_tanh(0.5f * x), 0.5f);
}

__global__ __launch_bounds__(32) void lstm_scan_wmma(
    const float* __restrict__ x,    // [T, B, I]
    const float* __restrict__ h0,   // [B, H]
    const float* __restrict__ c0,   // [B, H]
    const float* __restrict__ Wih,  // [4H, I]
    const float* __restrict__ Whh,  // [4H, H]
    const float* __restrict__ bih,  // [4H]
    const float* __restrict__ bhh,  // [4H]
    float* __restrict__ out)        // [T*B*H] ++ h[B*H] ++ c[B*H]
{
    __shared__ float gates[80 * 16];   // Gt[M][n]
    __shared__ float h_tile[HSZ * 16]; // h[j][n]

    const int lane  = threadIdx.x;  // 0..31
    const int half  = lane >> 4;    // 0 or 1
    const int lan16 = lane & 15;
    const int bbase = blockIdx.x * 16;
    const int koff  = 2 * half;

    // ---- loop-invariant A fragments (weights) ----
    v2f wfrag[5][8];
    #pragma unroll
    for (int mt = 0; mt < 5; ++mt) {
        const int m = 16 * mt + lan16;
        #pragma unroll
        for (int kk = 0; kk < 8; ++kk) {
            const int k0 = 4 * kk + koff;
            const int k1 = k0 + 1;
            float a0 = 0.0f, a1 = 0.0f;
            if (k0 < ISZ)             a0 = Wih[m * ISZ + k0];
            else if (k0 < ISZ + HSZ)  a0 = Whh[m * HSZ + (k0 - ISZ)];
            if (k1 < ISZ)             a1 = Wih[m * ISZ + k1];
            else if (k1 < ISZ + HSZ)  a1 = Whh[m * HSZ + (k1 - ISZ)];
            wfrag[mt][kk].x = a0;
            wfrag[mt][kk].y = a1;
        }
    }

    // ---- bias broadcast in C/D layout (acc initial value each step) ----
    float biasfrag[5][8];
    #pragma unroll
    for (int mt = 0; mt < 5; ++mt)
        #pragma unroll
        for (int v = 0; v < 8; ++v) {
            const int m = 16 * mt + v + 8 * half;
            biasfrag[mt][v] = bih[m] + bhh[m];
        }

    // ---- init: h tile in LDS, c in registers (10 (j,n) elements per lane) ----
    float c_reg[10];
    #pragma unroll
    for (int u = 0; u < 10; ++u) {
        const int e = lane + 32 * u;     // 0..319
        const int j = e >> 4;            // 0..19
        const int n = e & 15;
        h_tile[j * 16 + n] = h0[(bbase + n) * HSZ + j];
        c_reg[u]           = c0[(bbase + n) * HSZ + j];
    }
    __syncthreads();

    // ---- prime x-fragment pipeline (t = 0) ----
    const float* xr0 = x + ((size_t)0 * BATCH + bbase + lan16) * ISZ;
    float2 xv0 = *(const float2*)(xr0 + koff);
    float2 xv1 = *(const float2*)(xr0 + 4 + koff);
    float2 xv2 = make_float2(0.0f, 0.0f);
    if (half == 0) xv2 = *(const float2*)(xr0 + 8);

    for (int t = 0; t < T_STEPS; ++t) {
        // prefetch next timestep's x fragments (overlaps WMMA + elementwise)
        const int tn = (t + 1 < T_STEPS) ? (t + 1) : t;
        const float* xrn = x + ((size_t)tn * BATCH + bbase + lan16) * ISZ;
        float2 xn0 = *(const float2*)(xrn + koff);
        float2 xn1 = *(const float2*)(xrn + 4 + koff);
        float2 xn2 = make_float2(0.0f, 0.0f);
        if (half == 0) xn2 = *(const float2*)(xrn + 8);

        v8f acc[5];
        #pragma unroll
        for (int mt = 0; mt < 5; ++mt)
            #pragma unroll
            for (int v = 0; v < 8; ++v)
                acc[mt][v] = biasfrag[mt][v];

        #pragma unroll
        for (int kk = 0; kk < 8; ++kk) {
            v2f bfrag;
            if (kk == 0) {
                bfrag.x = xv0.x; bfrag.y = xv0.y;
            } else if (kk == 1) {
                bfrag.x = xv1.x; bfrag.y = xv1.y;
            } else if (kk == 2) {
                // half0: x K=8,9 ; half1: h j=0,1
                const float hb0 = h_tile[0 * 16 + lan16];
                const float hb1 = h_tile[1 * 16 + lan16];
                bfrag.x = half ? hb0 : xv2.x;
                bfrag.y = half ? hb1 : xv2.y;
            } else if (kk == 7) {
                // half0: h j=18,19 ; half1: zero pad (K=30,31)
                const float hb0 = h_tile[18 * 16 + lan16];
                const float hb1 = h_tile[19 * 16 + lan16];
                bfrag.x = half ? 0.0f : hb0;
                bfrag.y = half ? 0.0f : hb1;
            } else {
                const int j0 = 4 * kk - 10 + koff;   // kk=3..6 -> j0 in [2..17]
                bfrag.x = h_tile[(j0    ) * 16 + lan16];
                bfrag.y = h_tile[(j0 + 1) * 16 + lan16];
            }
            #pragma unroll
            for (int mt = 0; mt < 5; ++mt) {
                acc[mt] = __builtin_amdgcn_wmma_f32_16x16x4_f32(
                    false, wfrag[mt][kk], false, bfrag,
                    (short)0, acc[mt], false, false);
            }
        }

        // scatter gate pre-activations to LDS
        #pragma unroll
        for (int mt = 0; mt < 5; ++mt)
            #pragma unroll
            for (int v = 0; v < 8; ++v)
                gates[(16 * mt + v + 8 * half) * 16 + lan16] = acc[mt][v];
        __syncthreads();

        // elementwise LSTM cell update (PyTorch gate order i,f,g,o)
        #pragma unroll
        for (int u = 0; u < 10; ++u) {
            const int e = lane + 32 * u;
            const int j = e >> 4;
            const int n = e & 15;
            const float ig = act_sigmoid(gates[(          j) * 16 + n]);
            const float fg = act_sigmoid(gates[(    HSZ + j) * 16 + n]);
            const float gg = act_tanh   (gates[(2 * HSZ + j) * 16 + n]);
            const float og = act_sigmoid(gates[(3 * HSZ + j) * 16 + n]);
            const float cn = __builtin_fmaf(fg, c_reg[u], ig * gg);
            const float hn = og * act_tanh(cn);
            c_reg[u] = cn;
            h_tile[j * 16 + n] = hn;
        }
        __syncthreads();

        // coalesced output store: 32 consecutive dwords per store instruction
        float* out_t = out + (size_t)t * (BATCH * HSZ) + (size_t)bbase * HSZ;
        #pragma unroll
        for (int u = 0; u < 10; ++u) {
            const int i = lane + 32 * u;     // 0..319 flat within tile
            const int n = i / 20;
            const int j = i - n * 20;
            out_t[i] = h_tile[j * 16 + n];
        }

        xv0 = xn0; xv1 = xn1; xv2 = xn2;
    }

    // ---- final h (coalesced from LDS) ----
    float* hout = out + (size_t)T_STEPS * (BATCH * HSZ) + (size_t)bbase * HSZ;
    #pragma unroll
    for (int u = 0; u < 10; ++u) {
        const int i = lane + 32 * u;
        const int n = i / 20;
        const int j = i - n * 20;
        hout[i] = h_tile[j * 16 + n];
    }

    // ---- final c: stage through LDS (reuse gates buffer), then coalesced ----
    #pragma unroll
    for (int u = 0; u < 10; ++u) {
        const int e = lane + 32 * u;
        const int j = e >> 4;
        const int n = e & 15;
        gates[j * 16 + n] = c_reg[u];
    }
    __syncthreads();
    float* coutp = out + (size_t)T_STEPS * (BATCH * HSZ)
                       + (size_t)BATCH * HSZ + (size_t)bbase * HSZ;
    #pragma unroll
    for (int u = 0; u < 10; ++u) {
        const int i = lane + 32 * u;
        const int n = i / 20;
        const int j = i - n * 20;
        coutp[i] = gates[j * 16 + n];
    }
}

extern "C" void kernel_launch(void* const* d_in, const int* in_sizes, int n_in,
                              void* d_out, int out_size, void* d_ws, size_t ws_size,
                              hipStream_t stream) {
    (void)in_sizes; (void)n_in; (void)out_size; (void)d_ws; (void)ws_size;
    const float* x   = (const float*)d_in[0];
    const float* h0  = (const float*)d_in[1];
    const float* c0  = (const float*)d_in[2];
    const float* Wih = (const float*)d_in[3];
    const float* Whh = (const float*)d_in[4];
    const float* bih = (const float*)d_in[5];
    const float* bhh = (const float*)d_in[6];
    float* out = (float*)d_out;

    dim3 grid(BATCH / 16);  // 256 single-wave workgroups
    dim3 block(32);
    lstm_scan_wmma<<<grid, block, 0, stream>>>(x, h0, c0, Wih, Whh, bih, bhh, out);
}